// RSFPatchLoss2D_6519760355457
// MI455X (gfx1250) — compile-verified
//
#include <hip/hip_runtime.h>
#include <math.h>

typedef __attribute__((ext_vector_type(2))) float v2f;
typedef __attribute__((ext_vector_type(8))) float v8f;

#define LOG2E 1.4426950408889634f
#define LN2   0.6931471805599453f

#if __has_builtin(__builtin_amdgcn_tanhf)
#define USE_TANH 1
#endif

constexpr int NB  = 2;     // batch
constexpr int NC  = 3;     // channels
constexpr int HW  = 64;    // input H=W
constexpr int NN  = 2048;  // B * 32 * 32 output positions
constexpr int LF  = 128;   // filters
constexpr int KB  = 96;    // Bernstein order K

__global__ void zero_out(float* out) { out[0] = 0.f; }

// logC2[n] = (lgamma(K+1) - lgamma(n+1) - lgamma(K-n+1)) * log2(e)
__global__ void logc_kernel(float* __restrict__ logC2) {
  int n = threadIdx.x;
  if (n <= KB) {
    float v = lgammaf((float)(KB + 1)) - lgammaf((float)n + 1.f)
            - lgammaf((float)(KB - n) + 1.f);
    logC2[n] = v * LOG2E;
  }
}

// Zero-pad W[LF, Kreal] -> Wp[LF, Kpad] so the GEMM A-operand is unconditional.
__global__ void pad_w(const float* __restrict__ Wt, float* __restrict__ Wp,
                      int Kreal, int Kpad) {
  int idx = blockIdx.x * blockDim.x + threadIdx.x;
  if (idx >= LF * Kpad) return;
  int l = idx / Kpad, kk = idx % Kpad;
  Wp[idx] = (kk < Kreal) ? Wt[l * Kreal + kk] : 0.f;
}

// im2col in K-pair-interleaved layout: element (kidx, n) lives at
// col[(kidx>>1)*2*NN + 2*n + (kidx&1)].  One thread produces one K-pair as a
// single coalesced b64 store; pairs beyond Kreal are zeroed.
__global__ void im2col_pair(const float* __restrict__ Xg, const float* __restrict__ Xd,
                            float* __restrict__ colG, float* __restrict__ colD,
                            int Kreal, int ksz, int pad, int Kpairs) {
  int idx = blockIdx.x * blockDim.x + threadIdx.x;
  if (idx >= Kpairs * NN) return;
  const float* X = blockIdx.z ? Xd : Xg;
  float* col     = blockIdx.z ? colD : colG;
  int kp = idx / NN, n = idx % NN;
  int b = n >> 10, oh = (n >> 5) & 31, ow = n & 31;
  float vv[2] = {0.f, 0.f};
#pragma unroll
  for (int s = 0; s < 2; ++s) {
    int kidx = kp * 2 + s;
    if (kidx < Kreal) {
      int c  = kidx / (ksz * ksz), rem = kidx % (ksz * ksz);
      int kh = rem / ksz, kw = rem % ksz;
      int ih = oh * 2 - pad + kh, iw = ow * 2 - pad + kw;
      if (ih >= 0 && ih < HW && iw >= 0 && iw < HW)
        vv[s] = X[((b * NC + c) * HW + ih) * HW + iw];
    }
  }
  v2f out = {vv[0], vv[1]};
  *(v2f*)(col + (size_t)kp * 2 * NN + 2 * n) = out;
}

// proj[LF, NN] = Wp[LF, KPAD] x col[KPAD, NN] via V_WMMA_F32_16X16X4_F32.
// Wave tile 16(M) x 16(N); K fully unrolled (KPAD compile-time) -> KPAD/4 wmma
// per wave with clauseable b64 operand loads and no branches (EXEC all-ones).
// A 16x4 f32 layout: lanes 0-15 hold K={0,1}; lanes 16-31 hold K={2,3}.
// B 4x16 mirrored; C/D: VGPR v holds row M = v + 8*(lane>=16), col = lane&15.
template <int KPAD>
__global__ void wmma_gemm(const float* __restrict__ Wp,
                          const float* __restrict__ colG, const float* __restrict__ colD,
                          float* __restrict__ projX, float* __restrict__ projY) {
  const float* col = blockIdx.z ? colD : colG;
  float* proj      = blockIdx.z ? projY : projX;
  int lane = threadIdx.x & 31;
  int wave = threadIdx.x >> 5;
  int half = lane >> 4;     // 0: K{0,1}, 1: K{2,3}
  int lo   = lane & 15;
  int mBase = blockIdx.y * 16;
  int nBase = (blockIdx.x * 8 + wave) * 16;
  // A: pair (kk/2 + half) of row (mBase+lo)
  const v2f* arow = (const v2f*)(Wp + (size_t)(mBase + lo) * KPAD) + half;
  // B: pair-interleaved col; pair p at v2f index p*NN + n
  const v2f* bcol = (const v2f*)col + nBase + lo + (size_t)half * NN;
  v8f acc = {};
#pragma unroll
  for (int kk = 0; kk < KPAD; kk += 4) {
    v2f av = arow[kk >> 1];
    v2f bv = bcol[(size_t)(kk >> 1) * NN];
    acc = __builtin_amdgcn_wmma_f32_16x16x4_f32(
        /*neg_a=*/false, av, /*neg_b=*/false, bv,
        /*c_mod=*/(short)0, acc, /*reuse_a=*/false, /*reuse_b=*/false);
  }
  int row = mBase + half * 8;
#pragma unroll
  for (int v = 0; v < 8; ++v)
    proj[(size_t)(row + v) * NN + nBase + lo] = acc[v];
}

// invtau[l] = 1 / (tau_scale * (std_ddof1(projY[l,:]) + 1e-6) + 1e-4)
__global__ void tau_kernel(const float* __restrict__ projY,
                           float* __restrict__ invtau, float tau_scale) {
  __shared__ float s1[256], s2[256];
  int l = blockIdx.x, tid = threadIdx.x;
  float sum = 0.f, sq = 0.f;
  for (int i = tid; i < NN; i += 256) {
    float v = projY[(size_t)l * NN + i];
    sum += v; sq += v * v;
  }
  s1[tid] = sum; s2[tid] = sq;
  __syncthreads();
  for (int s = 128; s > 0; s >>= 1) {
    if (tid < s) { s1[tid] += s1[tid + s]; s2[tid] += s2[tid + s]; }
    __syncthreads();
  }
  if (tid == 0) {
    float mean = s1[0] / NN;
    float var  = (s2[0] - NN * mean * mean) / (NN - 1);
    float sd   = sqrtf(fmaxf(var, 0.f));
    invtau[l]  = 1.f / (tau_scale * (sd + 1e-6f) + 1e-4f);
  }
}

// Dominant kernel: U_i = mean_j sigmoid((x_i - y_j)/tau).
// sigmoid(t) = 0.5 + 0.5*tanh(t/2): accumulate tanh only -> 1 trans + 2 VALU
// per pair on gfx1250 (v_tanh_f32).  Fallback: exp2 + rcp (2 trans).
__global__ void u_kernel(const float* __restrict__ projX, const float* __restrict__ projY,
                         const float* __restrict__ invtau,
                         float* __restrict__ lu, float* __restrict__ l1u) {
  __shared__ float ys[256];
  int l = blockIdx.y, tid = threadIdx.x;
  int i = blockIdx.x * 256 + tid;
  float invt = invtau[l];
#ifdef USE_TANH
  float sc = invt * 0.5f;
#else
  float sc = invt;
#endif
  float xs  = projX[(size_t)l * NN + i] * sc;
  float acc = 0.f;
  for (int t = 0; t < NN; t += 256) {
    __syncthreads();
    ys[tid] = projY[(size_t)l * NN + t + tid] * sc;
    __syncthreads();
#pragma unroll 16
    for (int j = 0; j < 256; ++j) {
      float d = xs - ys[j];
#ifdef USE_TANH
      acc += __builtin_amdgcn_tanhf(d);
#else
      float e = __builtin_amdgcn_exp2f(-d * LOG2E);
      acc += __builtin_amdgcn_rcpf(1.f + e);
#endif
    }
  }
#ifdef USE_TANH
  float meanS = 0.5f + 0.5f * (acc * (1.f / NN));
#else
  float meanS = acc * (1.f / NN);
#endif
  float U = fminf(fmaxf(meanS, 1e-6f), 1.f - 1e-6f);
  lu [(size_t)l * NN + i] = __builtin_amdgcn_logf(U);        // log2
  l1u[(size_t)l * NN + i] = __builtin_amdgcn_logf(1.f - U);  // log2
}

// p_hat[n] = mean_i exp2(logC2[n] + n*log2U_i + (K-n)*log2(1-U_i)); then JS term.
__global__ void loss_kernel(const float* __restrict__ lu, const float* __restrict__ l1u,
                            const float* __restrict__ logC2, float* __restrict__ out) {
  __shared__ float slu[256], sl1u[256], red[128];
  int l = blockIdx.x, tid = threadIdx.x;
  float nf = (float)tid, kn = (float)(KB - tid);
  float lc = (tid <= KB) ? logC2[tid] : 0.f;
  float acc = 0.f;
  for (int t = 0; t < NN; t += 256) {
    __syncthreads();
    slu [tid]       = lu [(size_t)l * NN + t + tid];
    slu [tid + 128] = lu [(size_t)l * NN + t + tid + 128];
    sl1u[tid]       = l1u[(size_t)l * NN + t + tid];
    sl1u[tid + 128] = l1u[(size_t)l * NN + t + tid + 128];
    __syncthreads();
    if (tid <= KB) {
#pragma unroll 4
      for (int j = 0; j < 256; ++j)
        acc += __builtin_amdgcn_exp2f(lc + nf * slu[j] + kn * sl1u[j]);
    }
  }
  float contrib = 0.f;
  if (tid <= KB) {
    float p     = acc * (1.f / NN);
    float ratio = fmaxf(p * (float)(KB + 1), 1e-12f);   // p_hat / q
    float w     = 1.f / (1.f + ratio);
    float fjs   = 0.5f * (1.f + ratio) *
                  (LN2 + logf(fmaxf(w, 1e-12f)) + log1pf(-w));
    contrib = fjs * (1.f / (float)(KB + 1));            // q * f_js
  }
  red[tid] = contrib;
  __syncthreads();
  for (int s = 64; s > 0; s >>= 1) {
    if (tid < s) red[tid] += red[tid + s];
    __syncthreads();
  }
  if (tid == 0) atomicAdd(out, red[0] * (1.f / (3.f * (float)LF)));
}

extern "C" void kernel_launch(void* const* d_in, const int* in_sizes, int n_in,
                              void* d_out, int out_size, void* d_ws, size_t ws_size,
                              hipStream_t stream) {
  const float* Xg = (const float*)d_in[0];
  const float* Xd = (const float*)d_in[1];
  const float* Wb[3] = {(const float*)d_in[2], (const float*)d_in[3], (const float*)d_in[4]};
  float* out = (float*)d_out;
  float* ws  = (float*)d_ws;

  // Workspace layout (floats): ~6.7 MB total, L2-resident on MI455X.
  float* colG   = ws;                     // 148 * 2048 (pair-interleaved)
  float* colD   = colG   + 148 * NN;
  float* projX  = colD   + 148 * NN;      // 128 * 2048
  float* projY  = projX  + LF  * NN;
  float* lu     = projY  + LF  * NN;
  float* l1u    = lu     + LF  * NN;
  float* Wp     = l1u    + LF  * NN;      // 128 * 148 padded weights
  float* invtau = Wp     + LF  * 148;     // 128
  float* logC2  = invtau + LF;            // 97 (padded)

  const int   ksz[3] = {3, 5, 7};
  const float tsc[3] = {0.3f, 0.18f, 0.1f};

  zero_out<<<1, 1, 0, stream>>>(out);
  logc_kernel<<<1, 128, 0, stream>>>(logC2);

  for (int b = 0; b < 3; ++b) {
    int k = ksz[b], pad = k / 2;
    int Kreal  = NC * k * k;            // 27 / 75 / 147
    int Kpad   = (Kreal + 3) & ~3;      // 28 / 76 / 148
    int Kpairs = Kpad / 2;

    pad_w<<<(LF * Kpad + 255) / 256, 256, 0, stream>>>(Wb[b], Wp, Kreal, Kpad);
    im2col_pair<<<dim3((Kpairs * NN + 255) / 256, 1, 2), 256, 0, stream>>>(
        Xg, Xd, colG, colD, Kreal, k, pad, Kpairs);
    if (Kpad == 28)
      wmma_gemm<28><<<dim3(16, 8, 2), 256, 0, stream>>>(Wp, colG, colD, projX, projY);
    else if (Kpad == 76)
      wmma_gemm<76><<<dim3(16, 8, 2), 256, 0, stream>>>(Wp, colG, colD, projX, projY);
    else
      wmma_gemm<148><<<dim3(16, 8, 2), 256, 0, stream>>>(Wp, colG, colD, projX, projY);
    tau_kernel<<<128, 256, 0, stream>>>(projY, invtau, tsc[b]);
    u_kernel<<<dim3(8, 128), 256, 0, stream>>>(projX, projY, invtau, lu, l1u);
    loss_kernel<<<128, 128, 0, stream>>>(lu, l1u, logC2, out);
  }
}